// FPFH_29935922053528
// MI455X (gfx1250) — compile-verified
//
#include <hip/hip_runtime.h>
#include <math.h>

// ---------------- problem constants ----------------
#define NPTS   2048
#define BATCH  16
#define NBINS  11
#define FDIM   33          // 3*NBINS
#define R_N2   0.01f       // R_NORMAL^2
#define R_F2   0.04f       // R_FEATURE^2
#define CAP_N  30          // K_NORMAL
#define CAP_F  100         // K_FEATURE
#define QPB    16          // queries per block (one WMMA M tile)
#define NWAVES 8           // 256 threads
#define PI_F   3.14159265358979f

typedef __attribute__((ext_vector_type(2))) float    v2f;
typedef __attribute__((ext_vector_type(8))) float    v8f;
typedef __attribute__((ext_vector_type(4))) unsigned v4u;
typedef __attribute__((ext_vector_type(8))) unsigned v8u;

__device__ __forceinline__ float waveReduceSum(float v) {
  #pragma unroll
  for (int o = 16; o > 0; o >>= 1) v += __shfl_xor(v, o, 32);
  return v;
}

// ------------------------------------------------------------------
// Stage one batch's 2048 float4 candidates (32 KB) into LDS:
// path 1 — per-lane CDNA5 async copies (ASYNCcnt-tracked).
// Flat shared pointers carry the LDS offset in bits [31:0]
// (ISA 10.2 aperture rules), which is what VDST expects.
// ------------------------------------------------------------------
__device__ __forceinline__ void stage_candidates_async(const float4* __restrict__ gbase,
                                                       float4* cand) {
  for (int i = threadIdx.x; i < NPTS; i += NWAVES * 32) {
    const unsigned ldsAddr = (unsigned)(unsigned long long)(const void*)&cand[i];
    const float4* g = gbase + i;
    asm volatile("global_load_async_to_lds_b128 %0, %1, off"
                 :: "v"(ldsAddr), "v"(g) : "memory");
  }
  asm volatile("s_wait_asynccnt 0" ::: "memory");
  __syncthreads();
}

// ------------------------------------------------------------------
// path 2 — Tensor Data Mover: one descriptor, the DMA engine moves the
// whole 32 KB tile (8192 dwords as a 1-row tile), TENSORcnt-tracked.
// D# layout per CDNA5 ISA §8 (group0: count/lds/global/type=2;
// group1: data_size=4B, tensor_dim0=8192, tile_dim0=8192, tile_dim1=1).
// TDM ignores EXEC: issue from a single wave, then barrier.
// ------------------------------------------------------------------
__device__ __forceinline__ void stage_candidates_tdm(const float4* __restrict__ gbase,
                                                     float4* cand) {
  if ((threadIdx.x >> 5) == 0) {
    const unsigned long long ga = (unsigned long long)gbase;
    const unsigned lds = (unsigned)(unsigned long long)(const void*)cand;
    v4u g0;
    g0.x = 1u;                                                // count=1 (valid user D#)
    g0.y = lds;                                               // lds_addr (bytes)
    g0.z = (unsigned)ga;                                      // global_addr[31:0]
    g0.w = (unsigned)((ga >> 32) & 0x1FFFFFFu) | (2u << 30);  // addr[56:32] | type=2
    v8u g1;
    g1.s0 = (2u << 16);                 // wg_mask=0 | data_size=2 (4 bytes)
    g1.s1 = (8192u & 0xFFFFu) << 16;    // tensor_dim0[15:0] -> bits[63:48]
    g1.s2 = (8192u >> 16) | (1u << 16); // tensor_dim0[31:16] | tensor_dim1=1
    g1.s3 = (8192u & 0xFFFFu) << 16;    // tensor_dim1[31:16]=0 | tile_dim0=8192
    g1.s4 = 1u;                         // tile_dim1=1 | tile_dim2=0
    g1.s5 = 8192u;                      // tensor_dim0_stride[31:0]
    g1.s6 = 0u;                         // stride[47:32]=0 | dim1_stride[15:0]=0
    g1.s7 = 0u;
    asm volatile("tensor_load_to_lds %0, %1" :: "s"(g0), "s"(g1) : "memory");
    __builtin_amdgcn_s_wait_tensorcnt(0);
  }
  __syncthreads();
}

// ------------------------------------------------------------------
// Fused WMMA distance tile + radius-capped neighbor append.
// D = (-2*Q) * C^T + (|q|^2 + |c|^2)  via V_WMMA_F32_16X16X4_F32.
// A frag (16x4, M=query, K=coord):  lanes 0-15 -> (x,y), lanes 16-31 -> (z,0)
// B frag (4x16, K=coord, N=cand):   lanes 0-15 -> (x,y), lanes 16-31 -> (z,0)
// C/D (16x16): vgpr r, lane l -> M = r + 8*(l>=16), Ncol = l&15
// Candidates come from LDS (staged once per block).
// ------------------------------------------------------------------
template<int CAP>
__device__ __forceinline__ void dist_select(const float4* cand,   // LDS [2048]
                                            int qbase,
                                            const float* sqq,     // LDS [16]
                                            int* cnt,             // LDS [16]
                                            int* idxL,            // LDS [16*CAP]
                                            float* d2L,           // LDS [16*CAP]
                                            float r2, bool excludeSelf) {
  const int lane = threadIdx.x & 31;
  const int wave = threadIdx.x >> 5;
  const int nlo  = lane & 15;
  const bool hi  = lane >= 16;

  v2f afrag;
  {
    const float4 q = cand[qbase + nlo];
    afrag.x = hi ? (-2.0f * q.z) : (-2.0f * q.x);
    afrag.y = hi ? 0.0f          : (-2.0f * q.y);
  }

  for (int t = wave; t < NPTS / 16; t += NWAVES) {
    const int colbase = t * 16;
    const float4 c = cand[colbase + nlo];              // ds_load
    const float sqc = c.x * c.x + c.y * c.y + c.z * c.z;

    v2f bfrag;
    bfrag.x = hi ? c.z : c.x;
    bfrag.y = hi ? 0.0f : c.y;

    v8f cacc;
    #pragma unroll
    for (int r = 0; r < 8; r++) {
      const int m = r + (hi ? 8 : 0);
      cacc[r] = sqq[m] + sqc;
    }

    v8f d = __builtin_amdgcn_wmma_f32_16x16x4_f32(
        /*neg_a=*/false, afrag, /*neg_b=*/false, bfrag,
        /*c_mod=*/(short)0, cacc, /*reuse_a=*/false, /*reuse_b=*/false);

    // wave-uniform early-out: skip the append block for empty tiles
    float dmin = fminf(fminf(fminf(d[0], d[1]), fminf(d[2], d[3])),
                       fminf(fminf(d[4], d[5]), fminf(d[6], d[7])));
    if (__ballot(dmin <= r2) == 0ull) continue;

    const int col = colbase + nlo;
    #pragma unroll
    for (int r = 0; r < 8; r++) {
      const int m = r + (hi ? 8 : 0);
      const float d2 = fmaxf(d[r], 0.0f);
      bool ok = (d2 <= r2);
      if (excludeSelf && (col == qbase + m || d2 <= 0.0f)) ok = false;
      if (ok) {
        const int pos = atomicAdd(&cnt[m], 1);         // ds_add_rtn_u32
        if (pos < CAP) {
          idxL[m * CAP + pos] = col;
          d2L [m * CAP + pos] = d2;
        }
      }
    }
  }
}

// ---------------- kernel 0: pack (B,3,N) -> float4 AoS ----------------
__global__ void fpfh_pack(const float* __restrict__ xyz, float4* __restrict__ pts4) {
  const int n = blockIdx.x * blockDim.x + threadIdx.x;
  const int b = blockIdx.y;
  if (n >= NPTS) return;
  const float* src = xyz + (size_t)b * 3 * NPTS;
  pts4[(size_t)b * NPTS + n] = make_float4(src[n], src[NPTS + n], src[2 * NPTS + n], 0.0f);
}

// ---------------- analytic smallest eigenvector of sym 3x3 ----------------
__device__ __forceinline__ void smallest_evec(float a00, float a01, float a02,
                                              float a11, float a12, float a22,
                                              float& nx, float& ny, float& nz) {
  const float p1 = a01 * a01 + a02 * a02 + a12 * a12;
  const float qv = (a00 + a11 + a22) * (1.0f / 3.0f);
  const float p2 = (a00 - qv) * (a00 - qv) + (a11 - qv) * (a11 - qv) +
                   (a22 - qv) * (a22 - qv) + 2.0f * p1;
  const float p  = sqrtf(fmaxf(p2 * (1.0f / 6.0f), 1e-30f));
  const float ip = 1.0f / p;
  const float b00 = (a00 - qv) * ip, b01 = a01 * ip, b02 = a02 * ip;
  const float b11 = (a11 - qv) * ip, b12 = a12 * ip, b22 = (a22 - qv) * ip;
  const float detB = b00 * (b11 * b22 - b12 * b12)
                   - b01 * (b01 * b22 - b12 * b02)
                   + b02 * (b01 * b12 - b11 * b02);
  const float r   = fminf(fmaxf(0.5f * detB, -1.0f), 1.0f);
  const float phi = acosf(r) * (1.0f / 3.0f);
  const float lmin = qv + 2.0f * p * cosf(phi + 2.0943951023931953f); // +2pi/3

  const float r0x = a00 - lmin, r0y = a01, r0z = a02;
  const float r1x = a01, r1y = a11 - lmin, r1z = a12;
  const float r2x = a02, r2y = a12, r2z = a22 - lmin;

  float vx = r0y * r1z - r0z * r1y, vy = r0z * r1x - r0x * r1z, vz = r0x * r1y - r0y * r1x;
  float bn = vx * vx + vy * vy + vz * vz;
  {
    const float cx = r0y * r2z - r0z * r2y, cy = r0z * r2x - r0x * r2z, cz = r0x * r2y - r0y * r2x;
    const float n2 = cx * cx + cy * cy + cz * cz;
    if (n2 > bn) { vx = cx; vy = cy; vz = cz; bn = n2; }
  }
  {
    const float cx = r1y * r2z - r1z * r2y, cy = r1z * r2x - r1x * r2z, cz = r1x * r2y - r1y * r2x;
    const float n2 = cx * cx + cy * cy + cz * cz;
    if (n2 > bn) { vx = cx; vy = cy; vz = cz; bn = n2; }
  }
  if (bn < 1e-20f) { nx = 0.0f; ny = 0.0f; nz = 1.0f; return; }
  const float inv = rsqrtf(bn);
  nx = vx * inv; ny = vy * inv; nz = vz * inv;
  if (nz < 0.0f) { nx = -nx; ny = -ny; nz = -nz; }
}

// ---------------- kernel 1: normals ----------------
__global__ __launch_bounds__(256) void fpfh_normals(const float4* __restrict__ pts4,
                                                    float4* __restrict__ normals) {
  __shared__ float4 cand[NPTS];
  __shared__ float sqq[QPB];
  __shared__ int   cnt[QPB];
  __shared__ int   idxL[QPB * CAP_N];
  __shared__ float d2L[QPB * CAP_N];

  const int b = blockIdx.y, qbase = blockIdx.x * QPB, tid = threadIdx.x;
  stage_candidates_async(pts4 + (size_t)b * NPTS, cand);
  if (tid < QPB) {
    const float4 p = cand[qbase + tid];
    sqq[tid] = p.x * p.x + p.y * p.y + p.z * p.z;
    cnt[tid] = 0;
  }
  __syncthreads();
  dist_select<CAP_N>(cand, qbase, sqq, cnt, idxL, d2L, R_N2, /*excl*/false);
  __syncthreads();

  if (tid < QPB) {
    const int q = tid;
    const int m = min(cnt[q], CAP_N);
    const float inv = 1.0f / fmaxf((float)m, 1.0f);
    float cx = 0, cy = 0, cz = 0;
    for (int k = 0; k < m; k++) {
      const float4 p = cand[idxL[q * CAP_N + k]];
      cx += p.x; cy += p.y; cz += p.z;
    }
    cx *= inv; cy *= inv; cz *= inv;
    float c00 = 0, c01 = 0, c02 = 0, c11 = 0, c12 = 0, c22 = 0;
    for (int k = 0; k < m; k++) {
      const float4 p = cand[idxL[q * CAP_N + k]];
      const float dx = p.x - cx, dy = p.y - cy, dz = p.z - cz;
      c00 += dx * dx; c01 += dx * dy; c02 += dx * dz;
      c11 += dy * dy; c12 += dy * dz; c22 += dz * dz;
    }
    c00 *= inv; c01 *= inv; c02 *= inv; c11 *= inv; c12 *= inv; c22 *= inv;
    float nx, ny, nz;
    smallest_evec(c00, c01, c02, c11, c12, c22, nx, ny, nz);
    normals[(size_t)b * NPTS + qbase + q] = make_float4(nx, ny, nz, 0.0f);
  }
}

// ---------------- kernel 2: SPFH histograms ----------------
__global__ __launch_bounds__(256) void fpfh_spfh(const float4* __restrict__ pts4,
                                                 const float4* __restrict__ normals,
                                                 float* __restrict__ spfh) {
  __shared__ float4 cand[NPTS];
  __shared__ float sqq[QPB];
  __shared__ int   cnt[QPB];
  __shared__ int   idxL[QPB * CAP_F];
  __shared__ float d2L[QPB * CAP_F];
  __shared__ float hist[QPB * FDIM];

  const int b = blockIdx.y, qbase = blockIdx.x * QPB, tid = threadIdx.x;
  stage_candidates_tdm(pts4 + (size_t)b * NPTS, cand);
  if (tid < QPB) {
    const float4 p = cand[qbase + tid];
    sqq[tid] = p.x * p.x + p.y * p.y + p.z * p.z;
    cnt[tid] = 0;
  }
  for (int i = tid; i < QPB * FDIM; i += 256) hist[i] = 0.0f;
  __syncthreads();
  dist_select<CAP_F>(cand, qbase, sqq, cnt, idxL, d2L, R_F2, /*excl*/true);
  __syncthreads();

  const int lane = tid & 31, wave = tid >> 5;
  const float4* nrm = normals + (size_t)b * NPTS;

  for (int q = wave; q < QPB; q += NWAVES) {
    const int c = min(cnt[q], CAP_F);
    const float incr = 100.0f / fmaxf((float)c, 1.0f);
    const float4 pq = cand[qbase + q];
    const float4 nq = nrm[qbase + q];
    const float ux = nq.x, uy = nq.y, uz = nq.z;
    for (int k = lane; k < c; k += 32) {
      const int j = idxL[q * CAP_F + k];
      const float dist = sqrtf(fmaxf(d2L[q * CAP_F + k], 1e-12f));
      const float id = 1.0f / dist;
      const float4 pj = cand[j];
      const float4 nj = nrm[j];
      const float dvx = (pj.x - pq.x) * id, dvy = (pj.y - pq.y) * id, dvz = (pj.z - pq.z) * id;
      // v = cross(dvec, u), normalized
      float vx = dvy * uz - dvz * uy, vy = dvz * ux - dvx * uz, vz = dvx * uy - dvy * ux;
      const float ivn = 1.0f / fmaxf(sqrtf(vx * vx + vy * vy + vz * vz), 1e-12f);
      vx *= ivn; vy *= ivn; vz *= ivn;
      // wv = cross(u, v)
      const float wx = uy * vz - uz * vy, wy = uz * vx - ux * vz, wz = ux * vy - uy * vx;
      const float f1 = vx * nj.x + vy * nj.y + vz * nj.z;
      const float f2 = ux * dvx + uy * dvy + uz * dvz;
      const float f3 = atan2f(wx * nj.x + wy * nj.y + wz * nj.z,
                              ux * nj.x + uy * nj.y + uz * nj.z);
      int b1 = (int)floorf((float)NBINS * (f1 + 1.0f) * 0.5f);
      int b2 = (int)floorf((float)NBINS * (f2 + 1.0f) * 0.5f);
      int b3 = (int)floorf((float)NBINS * (f3 + PI_F) / (2.0f * PI_F));
      b1 = min(max(b1, 0), NBINS - 1);
      b2 = min(max(b2, 0), NBINS - 1);
      b3 = min(max(b3, 0), NBINS - 1);
      atomicAdd(&hist[q * FDIM + b1], incr);                 // ds_add_f32
      atomicAdd(&hist[q * FDIM + NBINS + b2], incr);
      atomicAdd(&hist[q * FDIM + 2 * NBINS + b3], incr);
    }
  }
  __syncthreads();
  for (int i = tid; i < QPB * FDIM; i += 256) {
    const int q = i / FDIM, f = i % FDIM;
    spfh[((size_t)b * NPTS + qbase + q) * FDIM + f] = hist[i];
  }
}

// ---------------- kernel 3: FPFH aggregation + normalize + transpose ----------------
__global__ __launch_bounds__(256) void fpfh_final(const float4* __restrict__ pts4,
                                                  const float* __restrict__ spfh,
                                                  float* __restrict__ out) {
  __shared__ float4 cand[NPTS];
  __shared__ float sqq[QPB];
  __shared__ int   cnt[QPB];
  __shared__ int   idxL[QPB * CAP_F];
  __shared__ float d2L[QPB * CAP_F];
  __shared__ float fpL[QPB * FDIM];

  const int b = blockIdx.y, qbase = blockIdx.x * QPB, tid = threadIdx.x;
  stage_candidates_tdm(pts4 + (size_t)b * NPTS, cand);
  if (tid < QPB) {
    const float4 p = cand[qbase + tid];
    sqq[tid] = p.x * p.x + p.y * p.y + p.z * p.z;
    cnt[tid] = 0;
  }
  __syncthreads();
  dist_select<CAP_F>(cand, qbase, sqq, cnt, idxL, d2L, R_F2, /*excl*/true);
  __syncthreads();

  const int lane = tid & 31, wave = tid >> 5;
  const float* sp = spfh + (size_t)b * NPTS * FDIM;

  for (int q = wave; q < QPB; q += NWAVES) {
    const int c = min(cnt[q], CAP_F);
    int   nidx[4];
    float nw[4];
    #pragma unroll
    for (int j = 0; j < 4; j++) {
      const int k = lane + j * 32;
      if (k < c) {
        nidx[j] = idxL[q * CAP_F + k];
        nw[j] = 1.0f / sqrtf(fmaxf(d2L[q * CAP_F + k], 1e-12f));   // mask/dist
      } else { nidx[j] = 0; nw[j] = 0.0f; }
    }
    for (int f = 0; f < FDIM; f++) {
      float acc = 0.0f;
      #pragma unroll
      for (int j = 0; j < 4; j++) acc += nw[j] * sp[(size_t)nidx[j] * FDIM + f];
      acc = waveReduceSum(acc);
      if (lane == 0) fpL[q * FDIM + f] = acc;
    }
  }
  __syncthreads();

  if (tid < QPB) {
    const int q = tid;
    const float* own = &sp[(size_t)(qbase + q) * FDIM];
    #pragma unroll
    for (int blk = 0; blk < 3; blk++) {
      float s = 0.0f;
      for (int f = 0; f < NBINS; f++) s += fpL[q * FDIM + blk * NBINS + f];
      const float scale = (s > 0.0f) ? (100.0f / fmaxf(s, 1e-12f)) : 1.0f;
      for (int f = 0; f < NBINS; f++) {
        const int fg = blk * NBINS + f;
        out[((size_t)b * FDIM + fg) * NPTS + qbase + q] =
            fpL[q * FDIM + fg] * scale + own[fg];
      }
    }
  }
}

// ---------------- host launcher ----------------
extern "C" void kernel_launch(void* const* d_in, const int* in_sizes, int n_in,
                              void* d_out, int out_size, void* d_ws, size_t ws_size,
                              hipStream_t stream) {
  (void)in_sizes; (void)n_in; (void)out_size; (void)ws_size;
  const float* xyz = (const float*)d_in[0];
  float* out = (float*)d_out;
  float* ws  = (float*)d_ws;

  float4* pts4    = (float4*)ws;                                   // 16*2048 float4
  float*  spfh    = ws + (size_t)BATCH * NPTS * 4;                 // 16*2048*33 floats
  float4* normals = (float4*)(ws + (size_t)BATCH * NPTS * 4
                                 + (size_t)BATCH * NPTS * FDIM);   // 16*2048 float4

  dim3 gp(NPTS / 256, BATCH);
  fpfh_pack<<<gp, 256, 0, stream>>>(xyz, pts4);

  dim3 g(NPTS / QPB, BATCH);
  fpfh_normals<<<g, 256, 0, stream>>>(pts4, normals);
  fpfh_spfh   <<<g, 256, 0, stream>>>(pts4, normals, spfh);
  fpfh_final  <<<g, 256, 0, stream>>>(pts4, spfh, out);
}